// BlockwiseToPixels_56882546868645
// MI455X (gfx1250) — compile-verified
//
#include <hip/hip_runtime.h>

typedef float v2f __attribute__((ext_vector_type(2)));
typedef float v8f __attribute__((ext_vector_type(8)));
typedef unsigned int uint_t;

#define NEXP   16
#define DDIM   256
#define PDIM   64
#define CHUNK  128     // tokens per workgroup (4 waves x 32 tokens)
#define DSTR   260     // padded LDS row stride (floats); 260 % 64 == 4 -> conflict-free b64

// ---------------- bucketing kernels ----------------

__global__ void k_zero(uint_t* counts) {
    if (threadIdx.x < NEXP) counts[threadIdx.x] = 0u;
}

__global__ void k_count(const int* __restrict__ idx, uint_t* __restrict__ counts, int n) {
    int i = blockIdx.x * blockDim.x + threadIdx.x;
    if (i < n) atomicAdd(&counts[idx[i] & (NEXP - 1)], 1u);
}

__global__ void k_prefix(const uint_t* __restrict__ counts,
                         uint_t* __restrict__ offsets,
                         uint_t* __restrict__ cursors) {
    uint_t run = 0;
    for (int e = 0; e < NEXP; ++e) {
        offsets[e] = run;
        cursors[e] = run;
        run += counts[e];
    }
    offsets[NEXP] = run;
}

__global__ void k_scatter(const int* __restrict__ idx, uint_t* __restrict__ cursors,
                          int* __restrict__ order, int n) {
    int i = blockIdx.x * blockDim.x + threadIdx.x;
    if (i < n) {
        uint_t p = atomicAdd(&cursors[idx[i] & (NEXP - 1)], 1u);
        order[p] = i;
    }
}

// ---------------- grouped GEMM with fp32 WMMA ----------------
//
// Per-wave tile: 32 tokens x 64 outputs, K=256 in steps of 4 via
// V_WMMA_F32_16X16X4_F32 (8 WMMA per K-step, each B frag reused 2x).
//
// A (16x4 f32): lane<16 -> M=lane,    K=k+{0,1}
//               lane>=16 -> M=lane-16, K=k+{2,3}
// B (4x16 f32): VGPR j: lane<16 -> K=k+j; lane>=16 -> K=k+j+2; N=lane&15 (+16n)
//   => lane needs WT[col][k+koff], WT[col][k+koff+1]: contiguous in the
//      transposed LDS copy -> single ds_load_b64 straight into the operand pair.

__global__ __launch_bounds__(128) void k_moe_gemm(
    const float* __restrict__ x, const float* __restrict__ W,
    const float* __restrict__ bias, const int* __restrict__ order,
    const uint_t* __restrict__ counts, const uint_t* __restrict__ offsets,
    float* __restrict__ out)
{
    __shared__ float sWT[PDIM * DSTR];   // W[e] transposed: [p][k], 65 KB

    // Map this workgroup -> (expert e, chunk within expert)
    const uint_t g = blockIdx.x;
    int e = -1; uint_t chunk = 0, acc = 0;
    for (int i = 0; i < NEXP; ++i) {
        uint_t nc = (counts[i] + (CHUNK - 1)) / CHUNK;
        if (e < 0 && g < acc + nc) { e = i; chunk = g - acc; }
        acc += nc;
    }
    if (e < 0) return;                   // surplus block
    const uint_t cnt   = counts[e];
    const uint_t ebase = offsets[e];

    // Stage W[e] ([256][64] row-major) transposed into LDS: sWT[p][d].
    {
        const float4* src = (const float4*)(W + (size_t)e * DDIM * PDIM);
        for (uint_t i = threadIdx.x; i < (DDIM * PDIM) / 4; i += 128u) {
            uint_t d  = i >> 4;          // row in W (k index)
            uint_t p4 = i & 15u;         // float4 group along p
            float4 v = src[i];
            sWT[(p4 * 4u + 0u) * DSTR + d] = v.x;
            sWT[(p4 * 4u + 1u) * DSTR + d] = v.y;
            sWT[(p4 * 4u + 2u) * DSTR + d] = v.z;
            sWT[(p4 * 4u + 3u) * DSTR + d] = v.w;
        }
    }
    __syncthreads();

    const uint_t wave = threadIdx.x >> 5;
    const uint_t lane = threadIdx.x & 31u;
    const uint_t mrow = lane & 15u;
    const uint_t hi   = lane >> 4;           // 0 | 1
    const uint_t koff = hi << 1;             // 0 | 2

    // Two 16-token A tiles per wave; clamp row for tails (stores are masked).
    const uint_t tileLi0 = chunk * CHUNK + wave * 32u;        // tile 0 rows 0..15
    const uint_t li0 = tileLi0 + mrow;
    const uint_t li1 = tileLi0 + 16u + mrow;
    const int tokA0 = order[ebase + ((li0 < cnt) ? li0 : 0u)];
    const int tokA1 = order[ebase + ((li1 < cnt) ? li1 : 0u)];
    const float* xr0 = x + (size_t)tokA0 * DDIM + koff;
    const float* xr1 = x + (size_t)tokA1 * DDIM + koff;

    // B fragment base: sWT[col][k+koff], col = 16n + mrow
    const float* bp0 = &sWT[( 0u + mrow) * DSTR + koff];
    const float* bp1 = &sWT[(16u + mrow) * DSTR + koff];
    const float* bp2 = &sWT[(32u + mrow) * DSTR + koff];
    const float* bp3 = &sWT[(48u + mrow) * DSTR + koff];

    v8f c00 = {}, c01 = {}, c02 = {}, c03 = {};   // tile 0
    v8f c10 = {}, c11 = {}, c12 = {}, c13 = {};   // tile 1

    #pragma unroll 2
    for (uint_t k = 0; k < DDIM; k += 4) {
        v2f a0 = *(const v2f*)(xr0 + k);          // global_load_b64, streams per row
        v2f a1 = *(const v2f*)(xr1 + k);
        v2f b0 = *(const v2f*)(bp0 + k);          // ds_load_b64, conflict-free
        v2f b1 = *(const v2f*)(bp1 + k);
        v2f b2 = *(const v2f*)(bp2 + k);
        v2f b3 = *(const v2f*)(bp3 + k);
        c00 = __builtin_amdgcn_wmma_f32_16x16x4_f32(false, a0, false, b0, (short)0, c00, false, false);
        c01 = __builtin_amdgcn_wmma_f32_16x16x4_f32(false, a0, false, b1, (short)0, c01, false, false);
        c02 = __builtin_amdgcn_wmma_f32_16x16x4_f32(false, a0, false, b2, (short)0, c02, false, false);
        c03 = __builtin_amdgcn_wmma_f32_16x16x4_f32(false, a0, false, b3, (short)0, c03, false, false);
        c10 = __builtin_amdgcn_wmma_f32_16x16x4_f32(false, a1, false, b0, (short)0, c10, false, false);
        c11 = __builtin_amdgcn_wmma_f32_16x16x4_f32(false, a1, false, b1, (short)0, c11, false, false);
        c12 = __builtin_amdgcn_wmma_f32_16x16x4_f32(false, a1, false, b2, (short)0, c12, false, false);
        c13 = __builtin_amdgcn_wmma_f32_16x16x4_f32(false, a1, false, b3, (short)0, c13, false, false);
    }

    // Bias (depends only on output column p = 16n + mrow).
    const float bv0 = bias[e * PDIM +  0 + mrow];
    const float bv1 = bias[e * PDIM + 16 + mrow];
    const float bv2 = bias[e * PDIM + 32 + mrow];
    const float bv3 = bias[e * PDIM + 48 + mrow];

    // C/D layout: VGPR v: lane<16 -> M=v, N=lane; lane>=16 -> M=v+8, N=lane-16.
    #pragma unroll
    for (int v = 0; v < 8; ++v) {
        uint_t M = (uint_t)v + (hi << 3);
        // tile 0
        uint_t liA = tileLi0 + M;
        if (liA < cnt) {
            int tok = order[ebase + liA];
            float* orow = out + (size_t)tok * PDIM;
            orow[ 0u + mrow] = c00[v] + bv0;
            orow[16u + mrow] = c01[v] + bv1;
            orow[32u + mrow] = c02[v] + bv2;
            orow[48u + mrow] = c03[v] + bv3;
        }
        // tile 1
        uint_t liB = tileLi0 + 16u + M;
        if (liB < cnt) {
            int tok = order[ebase + liB];
            float* orow = out + (size_t)tok * PDIM;
            orow[ 0u + mrow] = c10[v] + bv0;
            orow[16u + mrow] = c11[v] + bv1;
            orow[32u + mrow] = c12[v] + bv2;
            orow[48u + mrow] = c13[v] + bv3;
        }
    }
}

// ---------------- launcher ----------------

extern "C" void kernel_launch(void* const* d_in, const int* in_sizes, int n_in,
                              void* d_out, int out_size, void* d_ws, size_t ws_size,
                              hipStream_t stream) {
    const float* x    = (const float*)d_in[0];   // [B,T,D]
    const float* W    = (const float*)d_in[1];   // [E,D,P]
    const float* bias = (const float*)d_in[2];   // [E,P]
    const int*   idx  = (const int*)d_in[3];     // [B,T]
    float* out = (float*)d_out;                  // [B,T,P]

    const int ntok = in_sizes[3];                // B*T

    uint_t* wsw     = (uint_t*)d_ws;
    uint_t* counts  = wsw;                       // 16
    uint_t* offsets = wsw + 16;                  // 17
    uint_t* cursors = wsw + 33;                  // 16
    int*    order   = (int*)(wsw + 64);          // ntok ints

    k_zero<<<1, 32, 0, stream>>>(counts);
    k_count<<<(ntok + 255) / 256, 256, 0, stream>>>(idx, counts, ntok);
    k_prefix<<<1, 1, 0, stream>>>(counts, offsets, cursors);
    k_scatter<<<(ntok + 255) / 256, 256, 0, stream>>>(idx, cursors, order, ntok);

    const int nblocks = ntok / CHUNK + NEXP;     // covers sum_e ceil(cnt_e/CHUNK)
    k_moe_gemm<<<nblocks, 128, 0, stream>>>(x, W, bias, order, counts, offsets, out);

    (void)n_in; (void)out_size; (void)ws_size;
}